// Decoder_43301860278274
// MI455X (gfx1250) — compile-verified
//
#include <hip/hip_runtime.h>

// Decoder: fc1 -> fc2 (WMMA fp32, fused with conv0 linear) -> GCN conv0 -> ELU -> GCN conv1
// Target: MI455X gfx1250, wave32.

#define NG      4
#define BOT     64
#define FFN_    128
#define NPG     50000
#define NNODES  200000
#define NEDGE   3200000
#define NTILES  50000          // 800000 columns / 16
#define NCOLS   800000LL

typedef __attribute__((ext_vector_type(2))) float v2f;
typedef __attribute__((ext_vector_type(8))) float v8f;

__device__ __forceinline__ float elu1(float x) {
    return x > 0.f ? x : __expf(x) - 1.f;
}

// ---------------------------------------------------------------- K1: fc1
// z1 is [16][128], rows 4..15 zero-padded (WMMA A-matrix M padding).
__global__ void k_fc1(const float* __restrict__ x, const float* __restrict__ w,
                      const float* __restrict__ b, float* __restrict__ z1) {
    const int j = threadIdx.x;               // 0..127
    float a0 = 0.f, a1 = 0.f, a2 = 0.f, a3 = 0.f;
    for (int k = 0; k < BOT; ++k) {
        const float wv = w[k * FFN_ + j];
        a0 += x[0 * BOT + k] * wv;
        a1 += x[1 * BOT + k] * wv;
        a2 += x[2 * BOT + k] * wv;
        a3 += x[3 * BOT + k] * wv;
    }
    const float bj = b[j];
    z1[0 * FFN_ + j] = elu1(a0 + bj);
    z1[1 * FFN_ + j] = elu1(a1 + bj);
    z1[2 * FFN_ + j] = elu1(a2 + bj);
    z1[3 * FFN_ + j] = elu1(a3 + bj);
    for (int g = NG; g < 16; ++g) z1[g * FFN_ + j] = 0.f;
}

// ---------------------------------------------------- K2: fc2 via WMMA fp32
// One wave handles one 16-column tile per iteration: D(16x16) += A(16x128) * B(128x16)
// via 32 x V_WMMA_F32_16X16X4_F32. Tile j=16*t == features 0..15 of node t (per graph),
// so we immediately apply bias+ELU and the conv0 16->8 linear (via a per-wave LDS tile)
// and emit hw[node][0..7] without materializing h.
__global__ void __launch_bounds__(256)
k_fc2_wmma(const float* __restrict__ z1,     // [16][128] padded
           const float* __restrict__ w2,     // [128][800000]
           const float* __restrict__ b2,     // [800000]
           const float* __restrict__ w0,     // [16][8]
           float* __restrict__ hw) {         // [NNODES][8]
    __shared__ float lds[8][64];             // per-wave tile: [g*16 + c]

    const int lane  = threadIdx.x & 31;
    const int wave  = threadIdx.x >> 5;
    const int gwave = blockIdx.x * (blockDim.x >> 5) + wave;
    const int nwav  = gridDim.x * (blockDim.x >> 5);

    const int m  = lane & 15;                // M row (A) / N column (B)
    const int hi = lane >> 4;                // 0: K+{0,1}, 1: K+{2,3}

    // A matrix (16x128) held in registers across the whole tile loop.
    v2f A[32];
#pragma unroll
    for (int s = 0; s < 32; ++s) {
        const int k = 4 * s + 2 * hi;
        A[s].x = z1[m * FFN_ + k];
        A[s].y = z1[m * FFN_ + k + 1];
    }
    // conv0 weight column for this lane's output feature (lane&7).
    float w0c[16];
#pragma unroll
    for (int c = 0; c < 16; ++c) w0c[c] = w0[c * 8 + (lane & 7)];

    for (int t = gwave; t < NTILES; t += nwav) {
        const long long j = (long long)t * 16;
        v8f C = {};
#pragma unroll
        for (int s = 0; s < 32; ++s) {
            const int k = 4 * s + 2 * hi;
            v2f B;
            B.x = w2[(long long)k       * NCOLS + j + m];
            B.y = w2[(long long)(k + 1) * NCOLS + j + m];
            C = __builtin_amdgcn_wmma_f32_16x16x4_f32(
                    false, A[s], false, B, (short)0, C, false, false);
        }
        // C layout: VGPR v, lanes 0..15 -> row M=v, col N=lane. Rows 0..3 are graphs.
        const float bias = b2[j + m];
        if (hi == 0) {
#pragma unroll
            for (int g = 0; g < NG; ++g) {
                lds[wave][g * 16 + m] = elu1(C[g] + bias);
            }
        }
        asm volatile("" ::: "memory");       // keep ds_store before ds_load (same wave, in-order LDS)
        // 32 lanes -> (graph = lane>>3, out-feature = lane&7): conv0 linear 16->8.
        const int g2 = lane >> 3;
        const int o  = lane & 7;
        float acc = 0.f;
#pragma unroll
        for (int c = 0; c < 16; ++c) acc += lds[wave][g2 * 16 + c] * w0c[c];
        hw[((long long)(g2 * NPG + t)) * 8 + o] = acc;
        asm volatile("" ::: "memory");
    }
}

// ------------------------------------------------------------ degree kernels
__global__ void k_deg_init(float* __restrict__ deg) {
    const int n = blockIdx.x * blockDim.x + threadIdx.x;
    if (n < NNODES) deg[n] = 1.0f;           // self-loop weight
}

__global__ void k_deg_scatter(const long long* __restrict__ ei,
                              const float* __restrict__ ew,
                              float* __restrict__ deg) {
    const int e = blockIdx.x * blockDim.x + threadIdx.x;
    if (e >= NEDGE) return;
    const int c = (int)ei[NEDGE + e];        // col
    atomicAdd(deg + c, ew[e]);
}

// dinv = rsqrt(deg); init out0 with the self-loop message hw[n]*dinv^2.
__global__ void k_dinv_init0(const float* __restrict__ deg, float* __restrict__ dinv,
                             const float* __restrict__ hw, float* __restrict__ out0) {
    const int n = blockIdx.x * blockDim.x + threadIdx.x;
    if (n >= NNODES) return;
    const float di = rsqrtf(deg[n]);         // deg >= 1 always
    dinv[n] = di;
    const float s = di * di;
    const float4* p = (const float4*)(hw + (long long)n * 8);
    float4 a = p[0], b = p[1];
    a.x *= s; a.y *= s; a.z *= s; a.w *= s;
    b.x *= s; b.y *= s; b.z *= s; b.w *= s;
    float4* q = (float4*)(out0 + (long long)n * 8);
    q[0] = a; q[1] = b;
}

// -------------------------------------------------- conv0 edge scatter (8 feats)
__global__ void k_edge0(const long long* __restrict__ ei, const float* __restrict__ ew,
                        const float* __restrict__ dinv, const float* __restrict__ hw,
                        float* __restrict__ out0) {
    const int e = blockIdx.x * blockDim.x + threadIdx.x;
    if (e >= NEDGE) return;
    const int r = (int)ei[e];
    const int c = (int)ei[NEDGE + e];
    const float norm = dinv[r] * ew[e] * dinv[c];
    const float4* p = (const float4*)(hw + (long long)r * 8);
    const float4 a = p[0], b = p[1];
    float* o = out0 + (long long)c * 8;
    atomicAdd(o + 0, a.x * norm); atomicAdd(o + 1, a.y * norm);
    atomicAdd(o + 2, a.z * norm); atomicAdd(o + 3, a.w * norm);
    atomicAdd(o + 4, b.x * norm); atomicAdd(o + 5, b.y * norm);
    atomicAdd(o + 6, b.z * norm); atomicAdd(o + 7, b.w * norm);
}

// -------- per-node: conv0 bias + ELU, conv1 linear 8->1, init out with self-loop + b1
__global__ void k_y1(const float* __restrict__ out0, const float* __restrict__ b0,
                     const float* __restrict__ w1, const float* __restrict__ b1,
                     const float* __restrict__ dinv, float* __restrict__ y1,
                     float* __restrict__ out) {
    const int n = blockIdx.x * blockDim.x + threadIdx.x;
    if (n >= NNODES) return;
    const float4* p = (const float4*)(out0 + (long long)n * 8);
    const float4 u = p[0], v = p[1];
    const float vals[8] = {u.x, u.y, u.z, u.w, v.x, v.y, v.z, v.w};
    float acc = 0.f;
#pragma unroll
    for (int o = 0; o < 8; ++o) {
        float s = vals[o] + b0[o];
        s = elu1(s);
        acc += s * w1[o];
    }
    y1[n] = acc;
    const float di = dinv[n];
    out[n] = acc * di * di + b1[0];           // self-loop + conv1 bias
}

// ----------------------------------------------- conv1 edge scatter (1 feat)
__global__ void k_edge1(const long long* __restrict__ ei, const float* __restrict__ ew,
                        const float* __restrict__ dinv, const float* __restrict__ y1,
                        float* __restrict__ out) {
    const int e = blockIdx.x * blockDim.x + threadIdx.x;
    if (e >= NEDGE) return;
    const int r = (int)ei[e];
    const int c = (int)ei[NEDGE + e];
    const float norm = dinv[r] * ew[e] * dinv[c];
    atomicAdd(out + c, y1[r] * norm);
}

extern "C" void kernel_launch(void* const* d_in, const int* in_sizes, int n_in,
                              void* d_out, int out_size, void* d_ws, size_t ws_size,
                              hipStream_t stream) {
    const float*     x    = (const float*)d_in[0];
    const long long* ei   = (const long long*)d_in[1];   // int64 [2][E]
    const float*     ew   = (const float*)d_in[2];
    const float*     fc1w = (const float*)d_in[3];
    const float*     fc1b = (const float*)d_in[4];
    const float*     fc2w = (const float*)d_in[5];
    const float*     fc2b = (const float*)d_in[6];
    const float*     w0   = (const float*)d_in[7];
    const float*     b0   = (const float*)d_in[8];
    const float*     w1   = (const float*)d_in[9];
    const float*     b1   = (const float*)d_in[10];
    float* out = (float*)d_out;

    float* ws   = (float*)d_ws;
    float* z1   = ws;                        // 2048
    float* hw   = z1 + 2048;                 // NNODES*8
    float* deg  = hw + (size_t)NNODES * 8;   // NNODES
    float* dinv = deg + NNODES;              // NNODES
    float* out0 = dinv + NNODES;             // NNODES*8
    float* y1   = out0 + (size_t)NNODES * 8; // NNODES

    const int nodeBlocks = (NNODES + 255) / 256;
    const int edgeBlocks = NEDGE / 256;      // exact

    k_fc1<<<1, 128, 0, stream>>>(x, fc1w, fc1b, z1);
    k_deg_init<<<nodeBlocks, 256, 0, stream>>>(deg);
    k_deg_scatter<<<edgeBlocks, 256, 0, stream>>>(ei, ew, deg);
    k_fc2_wmma<<<1250, 256, 0, stream>>>(z1, fc2w, fc2b, w0, hw);   // 10000 waves, 5 tiles each
    k_dinv_init0<<<nodeBlocks, 256, 0, stream>>>(deg, dinv, hw, out0);
    k_edge0<<<edgeBlocks, 256, 0, stream>>>(ei, ew, dinv, hw, out0);
    k_y1<<<nodeBlocks, 256, 0, stream>>>(out0, b0, w1, b1, dinv, y1, out);
    k_edge1<<<edgeBlocks, 256, 0, stream>>>(ei, ew, dinv, y1, out);
}